// HyperGraphConv_14826227105922
// MI455X (gfx1250) — compile-verified
//
#include <hip/hip_runtime.h>

// HyperGraphConv for MI455X (gfx1250, wave32, WMMA).
// All matmuls use V_WMMA_F32_16X16X4_F32 (D = A(16x4) x B(4x16) + C).
// Compute-bound (~40 GFLOP f32) vs ~350MB traffic (~15us @ 23.3 TB/s).

#define B_   8
#define C_   96
#define N_   3136
#define M_   784
#define OUT_ 192
#define C4_  384
#define JCH  784    // column chunk for gram kernels (4 chunks of 49 tiles)

typedef __attribute__((ext_vector_type(2))) float v2f;
typedef __attribute__((ext_vector_type(8))) float v8f;

#define INF_F __builtin_inff()

static __device__ __forceinline__ v2f ld2(const float* p) { return *(const v2f*)p; }

static __device__ __forceinline__ v8f wmma4(v2f a, v2f b, v8f c) {
  // (neg_a, A, neg_b, B, c_mod, C, reuse_a, reuse_b)
  return __builtin_amdgcn_wmma_f32_16x16x4_f32(false, a, false, b, (short)0, c, false, false);
}

static __device__ __forceinline__ float wsum(float v) {
#pragma unroll
  for (int o = 16; o; o >>= 1) v += __shfl_xor(v, o, 32);
  return v;
}
static __device__ __forceinline__ float wmaxr(float v) {
#pragma unroll
  for (int o = 16; o; o >>= 1) v = fmaxf(v, __shfl_xor(v, o, 32));
  return v;
}
static __device__ __forceinline__ float wminr(float v) {
#pragma unroll
  for (int o = 16; o; o >>= 1) v = fminf(v, __shfl_xor(v, o, 32));
  return v;
}
static __device__ __forceinline__ void wargmin(float& v, int& i) {
#pragma unroll
  for (int o = 16; o; o >>= 1) {
    float ov = __shfl_xor(v, o, 32);
    int   oi = __shfl_xor(i, o, 32);
    if (ov < v || (ov == v && oi < i)) { v = ov; i = oi; }
  }
}
static __device__ __forceinline__ void wargmax(float& v, int& i) {
#pragma unroll
  for (int o = 16; o; o >>= 1) {
    float ov = __shfl_xor(v, o, 32);
    int   oi = __shfl_xor(i, o, 32);
    if (ov > v || (ov == v && oi < i)) { v = ov; i = oi; }
  }
}
// order-preserving float<->uint for atomicMax on floats
static __device__ __forceinline__ unsigned ford(float f) {
  unsigned u = __float_as_uint(f);
  return (u & 0x80000000u) ? ~u : (u | 0x80000000u);
}
static __device__ __forceinline__ float funord(unsigned u) {
  return __uint_as_float((u & 0x80000000u) ? (u & 0x7fffffffu) : ~u);
}

// ---------------- init ----------------
__global__ void k_init(float* mass, unsigned* dmaxu) {
  int t = blockIdx.x * blockDim.x + threadIdx.x;
  if (t < B_ * M_) mass[t] = 0.f;
  if (t < B_) dmaxu[t] = 0u;
}

// ---------------- (B,C,N) -> (B,N,C) + row norms ----------------
__global__ void k_transpose(const float* __restrict__ src, float* __restrict__ dst,
                            float* __restrict__ nrm) {
  int b = blockIdx.y;
  int n = blockIdx.x * blockDim.x + threadIdx.x;
  if (n >= N_) return;
  const float* s = src + (size_t)b * C_ * N_ + n;
  float* d = dst + ((size_t)b * N_ + n) * C_;
  float acc = 0.f;
  for (int c = 0; c < C_; c++) { float v = s[(size_t)c * N_]; d[c] = v; acc += v * v; }
  nrm[b * N_ + n] = acc;
}

// ---------------- density: top-16 smallest d per row + global dmax ----------------
__global__ void __launch_bounds__(256) k_density(const float* __restrict__ yt,
                                                 const float* __restrict__ ynorm,
                                                 const float* __restrict__ relpos,
                                                 float* __restrict__ density,
                                                 unsigned* __restrict__ dmaxu) {
  __shared__ float dch[16 * JCH];            // 50176 B; reused as argmin candidates at the end
  int b = blockIdx.y, i0 = blockIdx.x * 16;
  int tid = threadIdx.x, wave = tid >> 5, lane = tid & 31, r = lane & 15, hi = lane >> 4;
  const float* ytb = yt + (size_t)b * N_ * C_;

  v2f A[24];
  {
    const float* ap = ytb + (size_t)(i0 + r) * C_ + 2 * hi;
#pragma unroll
    for (int kk = 0; kk < 24; kk++) A[kk] = ld2(ap + 4 * kk);
  }
  float yni[8];
#pragma unroll
  for (int rr = 0; rr < 8; rr++) yni[rr] = ynorm[b * N_ + i0 + rr + 8 * hi];

  float t16[2][16];
  float rowmax[2] = {-INF_F, -INF_F};
#pragma unroll
  for (int q = 0; q < 2; q++)
#pragma unroll
    for (int k = 0; k < 16; k++) t16[q][k] = INF_F;

  for (int ch = 0; ch < 4; ch++) {
    int jbase = ch * JCH;
    for (int jt = wave; jt < 49; jt += 8) {          // uniform per wave -> EXEC all-1 at WMMA
      int j0 = jbase + jt * 16;
      v8f acc = {0.f, 0.f, 0.f, 0.f, 0.f, 0.f, 0.f, 0.f};
      const float* bp = ytb + (size_t)(j0 + r) * C_ + 2 * hi;
#pragma unroll
      for (int kk = 0; kk < 24; kk++) acc = wmma4(A[kk], ld2(bp + 4 * kk), acc);
      float ynj = ynorm[b * N_ + j0 + r];
#pragma unroll
      for (int rr = 0; rr < 8; rr++) {
        int M = rr + 8 * hi;
        float d = yni[rr] + ynj - 2.f * acc[rr] + relpos[(size_t)(i0 + M) * N_ + j0 + r];
        dch[M * JCH + jt * 16 + r] = d;
      }
    }
    __syncthreads();
#pragma unroll
    for (int q = 0; q < 2; q++) {                    // wave owns rows 2w, 2w+1
      int row = 2 * wave + q;
      const float* dr = &dch[row * JCH];
      float curmax = -INF_F; int amax = 0;
#pragma unroll
      for (int k = 0; k < 16; k++) if (t16[q][k] > curmax) { curmax = t16[q][k]; amax = k; }
      for (int c = lane; c < JCH; c += 32) {
        float v = dr[c];
        rowmax[q] = fmaxf(rowmax[q], v);
        if (v < curmax) {
#pragma unroll
          for (int k = 0; k < 16; k++) if (k == amax) t16[q][k] = v;
          curmax = -INF_F;
#pragma unroll
          for (int k = 0; k < 16; k++) if (t16[q][k] > curmax) { curmax = t16[q][k]; amax = k; }
        }
      }
    }
    __syncthreads();
  }
  // merge 32x16 lane-local candidates per row (lane-private LDS slots)
  float* cw = &dch[wave * 512];
#pragma unroll
  for (int q = 0; q < 2; q++) {
    int row = 2 * wave + q;
#pragma unroll
    for (int k = 0; k < 16; k++) cw[lane * 16 + k] = t16[q][k];
    float sum = 0.f;
    for (int it = 0; it < 16; it++) {
      float bv = INF_F; int bi = lane * 16;
#pragma unroll
      for (int k = 0; k < 16; k++) { float v = cw[lane * 16 + k]; if (v < bv) { bv = v; bi = lane * 16 + k; } }
      wargmin(bv, bi);
      sum += bv;
      if (lane == (bi >> 4)) cw[bi] = INF_F;         // only owner lane ever re-reads this slot
    }
    float rmax = wmaxr(rowmax[q]);
    if (lane == 0) {
      density[b * N_ + i0 + row] = __expf(-sum * (1.0f / 16.0f));
      atomicMax(&dmaxu[b], ford(rmax));
    }
  }
}

// ---------------- dist_peak * density = score ----------------
__global__ void __launch_bounds__(256) k_peak(const float* __restrict__ yt,
                                              const float* __restrict__ ynorm,
                                              const float* __restrict__ relpos,
                                              const float* __restrict__ density,
                                              const unsigned* __restrict__ dmaxu,
                                              float* __restrict__ score) {
  __shared__ float dch[16 * JCH];
  int b = blockIdx.y, i0 = blockIdx.x * 16;
  int tid = threadIdx.x, wave = tid >> 5, lane = tid & 31, r = lane & 15, hi = lane >> 4;
  const float* ytb = yt + (size_t)b * N_ * C_;
  float dmaxv = funord(dmaxu[b]);

  v2f A[24];
  {
    const float* ap = ytb + (size_t)(i0 + r) * C_ + 2 * hi;
#pragma unroll
    for (int kk = 0; kk < 24; kk++) A[kk] = ld2(ap + 4 * kk);
  }
  float yni[8];
#pragma unroll
  for (int rr = 0; rr < 8; rr++) yni[rr] = ynorm[b * N_ + i0 + rr + 8 * hi];

  float di[2], pmin[2];
#pragma unroll
  for (int q = 0; q < 2; q++) { di[q] = density[b * N_ + i0 + 2 * wave + q]; pmin[q] = dmaxv; }

  for (int ch = 0; ch < 4; ch++) {
    int jbase = ch * JCH;
    for (int jt = wave; jt < 49; jt += 8) {
      int j0 = jbase + jt * 16;
      v8f acc = {0.f, 0.f, 0.f, 0.f, 0.f, 0.f, 0.f, 0.f};
      const float* bp = ytb + (size_t)(j0 + r) * C_ + 2 * hi;
#pragma unroll
      for (int kk = 0; kk < 24; kk++) acc = wmma4(A[kk], ld2(bp + 4 * kk), acc);
      float ynj = ynorm[b * N_ + j0 + r];
#pragma unroll
      for (int rr = 0; rr < 8; rr++) {
        int M = rr + 8 * hi;
        float d = yni[rr] + ynj - 2.f * acc[rr] + relpos[(size_t)(i0 + M) * N_ + j0 + r];
        dch[M * JCH + jt * 16 + r] = d;
      }
    }
    __syncthreads();
#pragma unroll
    for (int q = 0; q < 2; q++) {
      int row = 2 * wave + q;
      const float* dr = &dch[row * JCH];
      for (int c = lane; c < JCH; c += 32) {
        float dj = density[b * N_ + jbase + c];
        if (dj > di[q]) pmin[q] = fminf(pmin[q], dr[c]);
      }
    }
    __syncthreads();
  }
#pragma unroll
  for (int q = 0; q < 2; q++) {
    float p = wminr(pmin[q]);
    if (lane == 0) score[b * N_ + i0 + 2 * wave + q] = p * di[q];
  }
}

// ---------------- top-784 by score (rank = descending, ties -> lower index) ----------------
__global__ void k_rank(const float* __restrict__ score, int* __restrict__ cidx) {
  int b = blockIdx.y;
  int i = blockIdx.x * blockDim.x + threadIdx.x;
  if (i >= N_) return;
  const float* s = score + b * N_;
  float si = s[i];
  int rank = 0;
  for (int j = 0; j < N_; j++) {
    float sj = s[j];
    rank += (sj > si) || (sj == si && j < i);
  }
  if (rank < M_) cidx[b * M_ + rank] = i;
}

__global__ void k_gather(const float* __restrict__ yt, const float* __restrict__ ynorm,
                         const int* __restrict__ cidx, float* __restrict__ cen,
                         float* __restrict__ cnorm) {
  int b = blockIdx.y, m = blockIdx.x, c = threadIdx.x;
  int idx = cidx[b * M_ + m];
  cen[((size_t)b * M_ + m) * C_ + c] = yt[((size_t)b * N_ + idx) * C_ + c];
  if (c == 0) cnorm[b * M_ + m] = ynorm[b * N_ + idx];
}

// ---------------- assign = softmax(-d(xf,cen)); fused mass + top-5 ----------------
__global__ void __launch_bounds__(256) k_assign(const float* __restrict__ xf,
                                                const float* __restrict__ xnorm,
                                                const float* __restrict__ cen,
                                                const float* __restrict__ cnorm,
                                                float* __restrict__ assign,
                                                float* __restrict__ mass,
                                                int* __restrict__ nnidx) {
  __shared__ float L[16 * M_];  // 50176 B
  int b = blockIdx.y, i0 = blockIdx.x * 16;
  int tid = threadIdx.x, wave = tid >> 5, lane = tid & 31, r = lane & 15, hi = lane >> 4;
  const float* xfb = xf + (size_t)b * N_ * C_;
  const float* cenb = cen + (size_t)b * M_ * C_;

  v2f A[24];
  {
    const float* ap = xfb + (size_t)(i0 + r) * C_ + 2 * hi;
#pragma unroll
    for (int kk = 0; kk < 24; kk++) A[kk] = ld2(ap + 4 * kk);
  }
  float xni[8];
#pragma unroll
  for (int rr = 0; rr < 8; rr++) xni[rr] = xnorm[b * N_ + i0 + rr + 8 * hi];

  for (int mt = wave; mt < 49; mt += 8) {
    int m0 = mt * 16;
    v8f acc = {0.f, 0.f, 0.f, 0.f, 0.f, 0.f, 0.f, 0.f};
    const float* bp = cenb + (size_t)(m0 + r) * C_ + 2 * hi;
#pragma unroll
    for (int kk = 0; kk < 24; kk++) acc = wmma4(A[kk], ld2(bp + 4 * kk), acc);
    float cnj = cnorm[b * M_ + m0 + r];
#pragma unroll
    for (int rr = 0; rr < 8; rr++) {
      int M = rr + 8 * hi;
      L[M * M_ + m0 + r] = 2.f * acc[rr] - xni[rr] - cnj;   // logits = -sq_dist
    }
  }
  __syncthreads();
  // softmax per row (wave owns rows 2w, 2w+1)
#pragma unroll
  for (int q = 0; q < 2; q++) {
    int row = 2 * wave + q;
    float* Lr = &L[row * M_];
    float mx = -INF_F;
    for (int m = lane; m < M_; m += 32) mx = fmaxf(mx, Lr[m]);
    mx = wmaxr(mx);
    float sm = 0.f;
    for (int m = lane; m < M_; m += 32) { float e = __expf(Lr[m] - mx); Lr[m] = e; sm += e; }
    sm = wsum(sm);
    float inv = 1.f / sm;
    float* arow = assign + (size_t)(b * N_ + i0 + row) * M_;
    for (int m = lane; m < M_; m += 32) { float a = Lr[m] * inv; Lr[m] = a; arow[m] = a; }
  }
  __syncthreads();
  // mass partial column-sums (before top-5 marking destroys L)
  for (int m = tid; m < M_; m += 256) {
    float s = 0.f;
#pragma unroll
    for (int row = 0; row < 16; row++) s += L[row * M_ + m];
    atomicAdd(&mass[b * M_ + m], s);
  }
  __syncthreads();
  // top-5 indices per row (descending, ties -> lower index)
#pragma unroll
  for (int q = 0; q < 2; q++) {
    int row = 2 * wave + q;
    float* Lr = &L[row * M_];
    int* nrow = nnidx + (size_t)(b * N_ + i0 + row) * 5;
    for (int t = 0; t < 5; t++) {
      float bv = -INF_F; int bi = 0x7fffffff;
      for (int m = lane; m < M_; m += 32) { float v = Lr[m]; if (v > bv) { bv = v; bi = m; } }
      wargmax(bv, bi);
      if (lane == 0) nrow[t] = bi;
      if (lane == (bi & 31)) Lr[bi] = -1.f;          // column is lane-private (m % 32 == lane)
    }
  }
}

// ---------------- agg = assign^T @ xf / mass ----------------
__global__ void __launch_bounds__(32) k_agg(const float* __restrict__ assign,
                                            const float* __restrict__ xf,
                                            const float* __restrict__ mass,
                                            float* __restrict__ agg) {
  int b = blockIdx.y, t = blockIdx.x;
  int m0 = (t / 6) * 16, c0 = (t % 6) * 16;
  int lane = threadIdx.x & 31, r = lane & 15, hi = lane >> 4;
  const float* asb = assign + (size_t)b * N_ * M_;
  const float* xfb = xf + (size_t)b * N_ * C_;
  v8f acc = {0.f, 0.f, 0.f, 0.f, 0.f, 0.f, 0.f, 0.f};
  for (int kk = 0; kk < N_ / 4; kk++) {
    int kb = 4 * kk + 2 * hi;
    v2f a, bb;
    a.x = asb[(size_t)kb * M_ + m0 + r];
    a.y = asb[(size_t)(kb + 1) * M_ + m0 + r];
    bb.x = xfb[(size_t)kb * C_ + c0 + r];
    bb.y = xfb[(size_t)(kb + 1) * C_ + c0 + r];
    acc = wmma4(a, bb, acc);
  }
#pragma unroll
  for (int rr = 0; rr < 8; rr++) {
    int m = m0 + rr + 8 * hi, c = c0 + r;
    agg[((size_t)b * M_ + m) * C_ + c] = acc[rr] / (mass[b * M_ + m] + 1e-6f);
  }
}

// ---------------- FFN (residual) + write refined ----------------
__global__ void __launch_bounds__(256) k_ffn(float* __restrict__ agg,
                                             const float* __restrict__ w1, const float* __restrict__ b1,
                                             const float* __restrict__ g1, const float* __restrict__ bt1,
                                             const float* __restrict__ w2, const float* __restrict__ b2,
                                             const float* __restrict__ g2, const float* __restrict__ bt2,
                                             float* __restrict__ refined) {
  __shared__ float H[16 * C4_];  // 24576 B
  int b = blockIdx.y, m0 = blockIdx.x * 16;
  int tid = threadIdx.x, wave = tid >> 5, lane = tid & 31, r = lane & 15, hi = lane >> 4;
  const float* aggb = agg + ((size_t)b * M_ + m0) * C_;
  v2f A[24];
  {
    const float* ap = aggb + (size_t)r * C_ + 2 * hi;
#pragma unroll
    for (int kk = 0; kk < 24; kk++) A[kk] = ld2(ap + 4 * kk);
  }
  for (int ot = wave; ot < 24; ot += 8) {
    int o0 = ot * 16;
    v8f acc = {0.f, 0.f, 0.f, 0.f, 0.f, 0.f, 0.f, 0.f};
    const float* wp = w1 + (size_t)(o0 + r) * C_ + 2 * hi;
#pragma unroll
    for (int kk = 0; kk < 24; kk++) acc = wmma4(A[kk], ld2(wp + 4 * kk), acc);
    float vb = b1[o0 + r], vg = g1[o0 + r], vt = bt1[o0 + r];
#pragma unroll
    for (int rr = 0; rr < 8; rr++) {
      int row = rr + 8 * hi;
      H[row * C4_ + o0 + r] = fmaxf((acc[rr] + vb) * vg + vt, 0.f);
    }
  }
  __syncthreads();
  if (wave < 6) {
    int c0 = wave * 16;
    v8f acc = {0.f, 0.f, 0.f, 0.f, 0.f, 0.f, 0.f, 0.f};
    for (int kk = 0; kk < 96; kk++) {
      int kb = 4 * kk + 2 * hi;
      v2f a; a.x = H[r * C4_ + kb]; a.y = H[r * C4_ + kb + 1];
      acc = wmma4(a, ld2(w2 + (size_t)(c0 + r) * C4_ + kb), acc);
    }
    float vb = b2[c0 + r], vg = g2[c0 + r], vt = bt2[c0 + r];
#pragma unroll
    for (int rr = 0; rr < 8; rr++) {
      int m = m0 + rr + 8 * hi, c = c0 + r;
      size_t ai = ((size_t)b * M_ + m) * C_ + c;
      float res = agg[ai] + (acc[rr] + vb) * vg + vt;
      agg[ai] = res;
      refined[((size_t)b * C_ + c) * M_ + m] = res;
    }
  }
}

// ---------------- gather top-5, build xcat, final 192x192 projection ----------------
__global__ void __launch_bounds__(256) k_final(const float* __restrict__ xf,
                                               const float* __restrict__ agg,
                                               const int* __restrict__ nnidx,
                                               const float* __restrict__ nnw,
                                               const float* __restrict__ nnb,
                                               const float* __restrict__ nng,
                                               const float* __restrict__ nnbt,
                                               float* __restrict__ out) {
  __shared__ float X[16 * OUT_];  // 12288 B
  int b = blockIdx.y, n0 = blockIdx.x * 16;
  int tid = threadIdx.x;
  for (int e = tid; e < 16 * C_; e += 256) {
    int row = e / C_, c = e % C_;
    int n = n0 + row;
    float xv = xf[((size_t)b * N_ + n) * C_ + c];
    const int* ni = nnidx + (size_t)(b * N_ + n) * 5;
    float mx = -INF_F;
#pragma unroll
    for (int t = 0; t < 5; t++) mx = fmaxf(mx, agg[((size_t)b * M_ + ni[t]) * C_ + c]);
    X[row * OUT_ + 2 * c] = xv;
    X[row * OUT_ + 2 * c + 1] = mx - xv;
  }
  __syncthreads();
  int wave = tid >> 5, lane = tid & 31, r = lane & 15, hi = lane >> 4;
  for (int ot = wave; ot < 12; ot += 8) {
    int o0 = ot * 16;
    v8f acc = {0.f, 0.f, 0.f, 0.f, 0.f, 0.f, 0.f, 0.f};
    const float* wp = nnw + (size_t)(o0 + r) * OUT_ + 2 * hi;
#pragma unroll
    for (int kk = 0; kk < 48; kk++) {
      v2f a; a.x = X[r * OUT_ + 4 * kk + 2 * hi]; a.y = X[r * OUT_ + 4 * kk + 2 * hi + 1];
      acc = wmma4(a, ld2(wp + 4 * kk), acc);
    }
    float vb = nnb[o0 + r], vg = nng[o0 + r], vt = nnbt[o0 + r];
#pragma unroll
    for (int rr = 0; rr < 8; rr++) {
      int nl = rr + 8 * hi, o = o0 + r;
      out[((size_t)b * OUT_ + o) * N_ + n0 + nl] = fmaxf((acc[rr] + vb) * vg + vt, 0.f);
    }
  }
}

extern "C" void kernel_launch(void* const* d_in, const int* in_sizes, int n_in,
                              void* d_out, int out_size, void* d_ws, size_t ws_size,
                              hipStream_t stream) {
  (void)in_sizes; (void)n_in; (void)out_size; (void)ws_size;
  const float* x    = (const float*)d_in[0];
  const float* relp = (const float*)d_in[1];
  const float* y    = (const float*)d_in[2];
  const float* w1   = (const float*)d_in[3];
  const float* b1   = (const float*)d_in[4];
  const float* g1   = (const float*)d_in[5];
  const float* bt1  = (const float*)d_in[6];
  const float* w2   = (const float*)d_in[7];
  const float* b2   = (const float*)d_in[8];
  const float* g2   = (const float*)d_in[9];
  const float* bt2  = (const float*)d_in[10];
  const float* nnw  = (const float*)d_in[11];
  const float* nnb  = (const float*)d_in[12];
  const float* nng  = (const float*)d_in[13];
  const float* nnbt = (const float*)d_in[14];

  float* out = (float*)d_out;
  float* refined = out + (size_t)B_ * OUT_ * N_;

  // workspace layout (~104 MB total)
  float* ws = (float*)d_ws;
  size_t o = 0;
  float* yt     = ws + o; o += (size_t)B_ * N_ * C_;
  float* xfp    = ws + o; o += (size_t)B_ * N_ * C_;
  float* ynorm  = ws + o; o += B_ * N_;
  float* xnorm  = ws + o; o += B_ * N_;
  float* dens   = ws + o; o += B_ * N_;
  float* score  = ws + o; o += B_ * N_;
  unsigned* dmaxu = (unsigned*)(ws + o); o += 64;
  int*   cidx   = (int*)(ws + o); o += B_ * M_;
  float* cen    = ws + o; o += (size_t)B_ * M_ * C_;
  float* cnorm  = ws + o; o += B_ * M_;
  float* mass   = ws + o; o += B_ * M_;
  int*   nnidx  = (int*)(ws + o); o += (size_t)B_ * N_ * 5;
  float* agg    = ws + o; o += (size_t)B_ * M_ * C_;
  float* assign = ws + o; o += (size_t)B_ * N_ * M_;

  dim3 tp((N_ + 255) / 256, B_);
  k_init<<<(B_ * M_ + 255) / 256, 256, 0, stream>>>(mass, dmaxu);
  k_transpose<<<tp, 256, 0, stream>>>(y, yt, ynorm);
  k_transpose<<<tp, 256, 0, stream>>>(x, xfp, xnorm);
  k_density<<<dim3(N_ / 16, B_), 256, 0, stream>>>(yt, ynorm, relp, dens, dmaxu);
  k_peak<<<dim3(N_ / 16, B_), 256, 0, stream>>>(yt, ynorm, relp, dens, dmaxu, score);
  k_rank<<<tp, 256, 0, stream>>>(score, cidx);
  k_gather<<<dim3(M_, B_), 96, 0, stream>>>(yt, ynorm, cidx, cen, cnorm);
  k_assign<<<dim3(N_ / 16, B_), 256, 0, stream>>>(xfp, xnorm, cen, cnorm, assign, mass, nnidx);
  k_agg<<<dim3((M_ / 16) * (C_ / 16), B_), 32, 0, stream>>>(assign, xfp, mass, agg);
  k_ffn<<<dim3(M_ / 16, B_), 256, 0, stream>>>(agg, w1, b1, g1, bt1, w2, b2, g2, bt2, refined);
  k_final<<<dim3(N_ / 16, B_), 256, 0, stream>>>(xfp, agg, nnidx, nnw, nnb, nng, nnbt, out);
}